// MPNNlayer_17420387352965
// MI455X (gfx1250) — compile-verified
//
#include <hip/hip_runtime.h>
#include <math.h>

#define Bdim 8
#define Ndim 256
#define ZDIM 64
#define EDIM 16
#define HDIM 64

typedef __attribute__((ext_vector_type(2))) float v2f;
typedef __attribute__((ext_vector_type(8))) float v8f;

// ---------------------------------------------------------------------------
// Kernel A: m_i[b,n,h] = z[b,n,:] . W_msg[h, 0:64]
//           m_j[b,n,h] = z[b,n,:] . W_msg[h, 64:128]
// ---------------------------------------------------------------------------
__global__ void mimj_kernel(const float* __restrict__ z,
                            const float* __restrict__ Wmsg,
                            float* __restrict__ mi, float* __restrict__ mj) {
    int bn = blockIdx.x;
    int h  = threadIdx.x;
    __shared__ float zrow[ZDIM];
    zrow[h] = z[bn * ZDIM + h];
    __syncthreads();
    const float* wi = Wmsg + h * (2 * ZDIM + EDIM);
    const float* wj = wi + ZDIM;
    float si = 0.f, sj = 0.f;
#pragma unroll 8
    for (int f = 0; f < ZDIM; ++f) {
        float zz = zrow[f];
        si += zz * wi[f];
        sj += zz * wj[f];
    }
    mi[bn * HDIM + h] = si;
    mj[bn * HDIM + h] = sj;
}

// ---------------------------------------------------------------------------
// Kernel B (dominant): per (b,i) workgroup:
//   D[j,h]  = e_feat[b,i,j,:] . W_e[h,:] + m_j[b,j,h]   (fp32 WMMA, C = mj tile)
//   agg[h]  = max_j( D[j,h] + maskF[j] ) + m_i[b,i,h] + b_msg[h]
//   maskF[j] = adj==1 ? 0 : -inf  (additive mask, branch-free)
// 256 threads = 8 waves; wave w owns j-rows [32w, 32w+32).
// ---------------------------------------------------------------------------
#define E_STRIDE 20   // 16 cols padded to 20 floats: conflict-free & 16B aligned
#define W_STRIDE 80   // 64 cols padded to 80: disjoint banks for the 2 K-halves

__global__ void msg_kernel(const float* __restrict__ e_feat,
                           const int*   __restrict__ adj,
                           const float* __restrict__ Wmsg,
                           const float* __restrict__ bmsg,
                           const float* __restrict__ mj,
                           const float* __restrict__ mi,
                           float* __restrict__ agg) {
    __shared__ float E[Ndim * E_STRIDE];     // e_feat[b,i,:,:]  (20.0 KB)
    __shared__ float WeT[EDIM * W_STRIDE];   // WeT[e][h] = W_msg[h,128+e] (5 KB)
    __shared__ float maskF[Ndim];            // 0 / -inf additive mask (1 KB)
    __shared__ float red[8 * HDIM];          // per-wave partial maxima (2 KB)

    const int bn = blockIdx.x;        // b*256 + i
    const int b  = bn >> 8;
    const int t  = threadIdx.x;       // 0..255

    const float NEG_INF = -__builtin_inff();

    // ---- stage e_feat row t (16 floats = 4 float4, coalesced) ----
    {
        const float4* src = (const float4*)(e_feat + (size_t)bn * Ndim * EDIM + t * EDIM);
        float4* dst = (float4*)(E + t * E_STRIDE);
#pragma unroll
        for (int q = 0; q < 4; ++q) dst[q] = src[q];
    }
    // ---- stage W_e transposed: WeT[e*W_STRIDE + h] ----
    {
#pragma unroll
        for (int q = 0; q < 4; ++q) {
            int idx = t * 4 + q;              // 0..1023
            int e = idx >> 6, h = idx & 63;
            WeT[e * W_STRIDE + h] = Wmsg[h * (2 * ZDIM + EDIM) + 2 * ZDIM + e];
        }
    }
    maskF[t] = (adj[(size_t)bn * Ndim + t] == 1) ? 0.f : NEG_INF;
    __syncthreads();

    const int wave  = t >> 5;
    const int lane  = t & 31;
    const int lm    = lane & 15;      // M/N position within half-wave
    const int hi    = lane >> 4;      // 0: lanes 0-15, 1: lanes 16-31
    const int jwave = wave * 32;

    // per-lane additive mask registers: j = jwave + jt*16 + r + 8*hi
    float mk[16];
#pragma unroll
    for (int jt = 0; jt < 2; ++jt)
#pragma unroll
        for (int r = 0; r < 8; ++r)
            mk[jt * 8 + r] = maskF[jwave + jt * 16 + r + 8 * hi];

    float vm0 = NEG_INF, vm1 = NEG_INF, vm2 = NEG_INF, vm3 = NEG_INF;
    float vmarr[4];

#pragma unroll
    for (int jt = 0; jt < 2; ++jt) {
        const int jbase = jwave + jt * 16;

        // A-frags for all 4 K-steps (ht-invariant): lanes0-15 K=k0+0/1,
        // lanes16-31 K=k0+2/3, row M = jbase+lm
        v2f afr[4];
#pragma unroll
        for (int k = 0; k < 4; ++k) {
            const int ar = (jbase + lm) * E_STRIDE + 4 * k + 2 * hi;
            afr[k].x = E[ar];
            afr[k].y = E[ar + 1];
        }

        float vmn[4] = {NEG_INF, NEG_INF, NEG_INF, NEG_INF};
#pragma unroll
        for (int ht = 0; ht < 4; ++ht) {
            const int hbase = ht * 16;

            // C-init = mj tile (D layout: VGPR r -> M = r + 8*hi, N = lm).
            // 8 unconditional global_load_b32 w/ immediate offsets; L2 hits.
            const float* mjp = mj + ((size_t)b * Ndim + jbase + 8 * hi) * HDIM
                                  + hbase + lm;
            v8f acc;
#pragma unroll
            for (int r = 0; r < 8; ++r) acc[r] = mjp[r * HDIM];

#pragma unroll
            for (int k = 0; k < 4; ++k) {
                const int k0 = 4 * k;
                v2f bm;
                const int be = (k0 + 2 * hi) * W_STRIDE + hbase + lm;
                bm.x = WeT[be];
                bm.y = WeT[be + W_STRIDE];
                acc = __builtin_amdgcn_wmma_f32_16x16x4_f32(
                    false, afr[k], false, bm, (short)0, acc, false, false);
            }

            // branch-free masked max: acc already includes mj
            float v = vmn[ht];
#pragma unroll
            for (int r = 0; r < 8; ++r) v = fmaxf(v, acc[r] + mk[jt * 8 + r]);
            vmn[ht] = v;
        }
        vm0 = fmaxf(vm0, vmn[0]);
        vm1 = fmaxf(vm1, vmn[1]);
        vm2 = fmaxf(vm2, vmn[2]);
        vm3 = fmaxf(vm3, vmn[3]);
    }
    vmarr[0] = vm0; vmarr[1] = vm1; vmarr[2] = vm2; vmarr[3] = vm3;

#pragma unroll
    for (int ht = 0; ht < 4; ++ht) {
        // combine the two half-wave j-sets (same h columns)
        float v = fmaxf(vmarr[ht], __shfl_xor(vmarr[ht], 16, 32));
        if (lane < 16) red[wave * HDIM + ht * 16 + lm] = v;
    }
    __syncthreads();

    if (t < HDIM) {
        float m = red[t];
#pragma unroll
        for (int w = 1; w < 8; ++w) m = fmaxf(m, red[w * HDIM + t]);
        agg[(size_t)bn * HDIM + t] = m + mi[(size_t)bn * HDIM + t] + bmsg[t];
    }
}

// ---------------------------------------------------------------------------
// Kernel C: out[b,n,h] = [z | agg][b,n,:] . W_upd[h,:] + b_upd[h]
// ---------------------------------------------------------------------------
__global__ void upd_kernel(const float* __restrict__ z,
                           const float* __restrict__ agg,
                           const float* __restrict__ Wupd,
                           const float* __restrict__ bupd,
                           float* __restrict__ out) {
    int bn = blockIdx.x;
    int h  = threadIdx.x;
    __shared__ float feat[ZDIM + HDIM];
    feat[h]        = z[bn * ZDIM + h];
    feat[ZDIM + h] = agg[bn * HDIM + h];
    __syncthreads();
    const float* w = Wupd + h * (ZDIM + HDIM);
    float s = bupd[h];
#pragma unroll 8
    for (int f = 0; f < ZDIM + HDIM; ++f) s += feat[f] * w[f];
    out[bn * HDIM + h] = s;
}

extern "C" void kernel_launch(void* const* d_in, const int* in_sizes, int n_in,
                              void* d_out, int out_size, void* d_ws, size_t ws_size,
                              hipStream_t stream) {
    const float* z      = (const float*)d_in[0];
    const float* e_feat = (const float*)d_in[1];
    const int*   adj    = (const int*)d_in[2];
    const float* W_msg  = (const float*)d_in[3];
    const float* b_msg  = (const float*)d_in[4];
    const float* W_upd  = (const float*)d_in[5];
    const float* b_upd  = (const float*)d_in[6];
    float*       out    = (float*)d_out;

    const int BN = Bdim * Ndim;               // 2048
    float* mi  = (float*)d_ws;                // BN*64 floats
    float* mj  = mi + BN * HDIM;              // BN*64 floats
    float* agg = mj + BN * HDIM;              // BN*64 floats  (total 1.5 MB)

    mimj_kernel<<<BN, HDIM, 0, stream>>>(z, W_msg, mi, mj);
    msg_kernel <<<BN, 256,  0, stream>>>(e_feat, adj, W_msg, b_msg, mj, mi, agg);
    upd_kernel <<<BN, HDIM, 0, stream>>>(z, agg, W_upd, b_upd, out);
}